// TrajectoryModel_73813307949241
// MI455X (gfx1250) — compile-verified
//
#include <hip/hip_runtime.h>
#include <math.h>

#define HD    64     // hidden dim H
#define LSEQ  1024   // sequence length L
#define BATCH 64     // batch B
#define VOC   32000  // vocab V

typedef float v2f __attribute__((ext_vector_type(2)));
typedef float v8f __attribute__((ext_vector_type(8)));

// fp32 WMMA: D(16x16) = A(16x4) * B(4x16) + C, wave32
__device__ __forceinline__ v8f wmma_f32x4(v2f a, v2f b, v8f c) {
  return __builtin_amdgcn_wmma_f32_16x16x4_f32(
      /*neg_a=*/false, a, /*neg_b=*/false, b,
      /*c_mod=*/(short)0, c, /*reuse_a=*/false, /*reuse_b=*/false);
}

// ---- CDNA5 async global->LDS copy (ASYNCcnt path), with safe fallback ----
#if __has_builtin(__builtin_amdgcn_global_load_async_to_lds_b32)
#define ASYNC_LDS 1
#else
#define ASYNC_LDS 0
#endif

typedef __attribute__((address_space(1))) int global_i32;
typedef __attribute__((address_space(3))) int lds_i32;

__device__ __forceinline__ void async_copy_b32(const float* g, float* l) {
#if ASYNC_LDS
  __builtin_amdgcn_global_load_async_to_lds_b32(
      (global_i32*)g, (lds_i32*)l, /*offset=*/0, /*cpol=*/0);
#else
  *l = *g;
#endif
}

__device__ __forceinline__ void wait_async0() {
#if ASYNC_LDS
#if __has_builtin(__builtin_amdgcn_s_wait_asynccnt)
  __builtin_amdgcn_s_wait_asynccnt(0);
#else
  asm volatile("s_wait_asynccnt 0x0" ::: "memory");
#endif
#endif
}

// ---------------------------------------------------------------------------
// Kernel 1: per-token pipeline, 16 tokens per wave (blockDim = 32).
//   h = embed[seq]; hid = relu(h@W1+b1); x = h + hid@W2+b2;
//   xn = LN(x)*gamma+beta; k_all = xn @ kpW
// All three GEMMs via v_wmma_f32_16x16x4_f32; gather via async-to-LDS.
// ---------------------------------------------------------------------------
__global__ __launch_bounds__(32) void token_prep_kernel(
    const int* __restrict__ seq, const float* __restrict__ embed,
    const float* __restrict__ W1, const float* __restrict__ b1,
    const float* __restrict__ W2, const float* __restrict__ b2,
    const float* __restrict__ gamma, const float* __restrict__ beta,
    const float* __restrict__ kpW, float* __restrict__ k_all)
{
  __shared__ float h_lds[16][HD];        // 4 KB
  __shared__ float hid_lds[16][2 * HD];  // 8 KB
  __shared__ float x_lds[16][HD];        // 4 KB
  __shared__ float mu_s[16];
  __shared__ float rs_s[16];

  const int lane = threadIdx.x;       // 0..31
  const int half = lane >> 4;         // 0 | 1
  const int l16  = lane & 15;
  const int tg0  = blockIdx.x * 16;   // first flat token (b*L + l)

  // --- embedding gather: 16 tokens x 64 cols, fire-and-forget into LDS ---
  for (int idx = lane; idx < 16 * HD; idx += 32) {
    const int r = idx >> 6, c = idx & 63;
    const int tok = seq[tg0 + r];
    async_copy_b32(&embed[(size_t)tok * HD + c], &h_lds[r][c]);
  }
  wait_async0();
  __syncthreads();

  // --- GEMM1: hidden = relu(h @ W1 + b1), 16 x 128 ---
  for (int nt = 0; nt < 8; ++nt) {
    const int n0 = nt * 16;
    const float bias = b1[n0 + l16];
    v8f acc;
    #pragma unroll
    for (int v = 0; v < 8; ++v) acc[v] = bias;
    #pragma unroll
    for (int k0 = 0; k0 < HD; k0 += 4) {
      const int ka = k0 + 2 * half;
      v2f a, b;
      a.x = h_lds[l16][ka];
      a.y = h_lds[l16][ka + 1];
      b.x = W1[ka * (2 * HD) + n0 + l16];
      b.y = W1[(ka + 1) * (2 * HD) + n0 + l16];
      acc = wmma_f32x4(a, b, acc);
    }
    #pragma unroll
    for (int v = 0; v < 8; ++v)
      hid_lds[v + 8 * half][n0 + l16] = fmaxf(acc[v], 0.0f);
  }
  __syncthreads();

  // --- GEMM2: x = h + hidden @ W2 + b2, 16 x 64 ---
  for (int nt = 0; nt < 4; ++nt) {
    const int n0 = nt * 16;
    const float bias = b2[n0 + l16];
    v8f acc;
    #pragma unroll
    for (int v = 0; v < 8; ++v) acc[v] = bias;
    #pragma unroll
    for (int k0 = 0; k0 < 2 * HD; k0 += 4) {
      const int ka = k0 + 2 * half;
      v2f a, b;
      a.x = hid_lds[l16][ka];
      a.y = hid_lds[l16][ka + 1];
      b.x = W2[ka * HD + n0 + l16];
      b.y = W2[(ka + 1) * HD + n0 + l16];
      acc = wmma_f32x4(a, b, acc);
    }
    #pragma unroll
    for (int v = 0; v < 8; ++v) {
      const int m = v + 8 * half;
      x_lds[m][n0 + l16] = acc[v] + h_lds[m][n0 + l16];
    }
  }
  __syncthreads();

  // --- LayerNorm stats (one row per lane, lanes 0..15) ---
  if (lane < 16) {
    float s = 0.0f, s2 = 0.0f;
    #pragma unroll
    for (int c = 0; c < HD; ++c) {
      const float x = x_lds[lane][c];
      s += x; s2 += x * x;
    }
    const float mu  = s * (1.0f / HD);
    const float var = s2 * (1.0f / HD) - mu * mu;
    mu_s[lane] = mu;
    rs_s[lane] = rsqrtf(var + 1e-5f);
  }
  __syncthreads();

  // --- normalize in place ---
  for (int idx = lane; idx < 16 * HD; idx += 32) {
    const int r = idx >> 6, c = idx & 63;
    x_lds[r][c] = (x_lds[r][c] - mu_s[r]) * rs_s[r] * gamma[c] + beta[c];
  }
  __syncthreads();

  // --- GEMM3: k = xn @ kpW, 16 x 64, store to global k_all ---
  for (int nt = 0; nt < 4; ++nt) {
    const int n0 = nt * 16;
    v8f acc;
    #pragma unroll
    for (int v = 0; v < 8; ++v) acc[v] = 0.0f;
    #pragma unroll
    for (int k0 = 0; k0 < HD; k0 += 4) {
      const int ka = k0 + 2 * half;
      v2f a, b;
      a.x = x_lds[l16][ka];
      a.y = x_lds[l16][ka + 1];
      b.x = kpW[ka * HD + n0 + l16];
      b.y = kpW[(ka + 1) * HD + n0 + l16];
      acc = wmma_f32x4(a, b, acc);
    }
    #pragma unroll
    for (int v = 0; v < 8; ++v) {
      const int m = v + 8 * half;
      k_all[(size_t)(tg0 + m) * HD + n0 + l16] = acc[v];
    }
  }
}

// ---------------------------------------------------------------------------
// Kernel 2: sequential fast-weight scan, one block (64 threads) per batch.
// Thread i keeps row i of M entirely in registers (unrolled).
// Next k-row is async-prefetched into an LDS double buffer so the global
// load latency is off the sequential critical path.
// ---------------------------------------------------------------------------
__global__ __launch_bounds__(64) void scan_kernel(
    const float* __restrict__ k_all, const float* __restrict__ rpW,
    const float* __restrict__ rpb, float* __restrict__ proj)
{
  const int b = blockIdx.x;
  const int i = threadIdx.x;  // row 0..63
  const float* kb = k_all + (size_t)b * LSEQ * HD;

  float M[HD];
  #pragma unroll
  for (int j = 0; j < HD; ++j) M[j] = 0.0f;

  __shared__ float kbuf[2][HD];   // double-buffered k rows
  __shared__ float kn_s[HD];
  __shared__ float red[HD];

  // prologue: stage row 0
  async_copy_b32(kb + i, &kbuf[0][i]);
  wait_async0();
  __syncthreads();

  for (int t = 0; t < LSEQ - 1; ++t) {
    const int cur = t & 1;
    const int nxt = cur ^ 1;
    // prefetch row t+1 (rows 1..L-1; the last one is q)
    async_copy_b32(kb + (size_t)(t + 1) * HD + i, &kbuf[nxt][i]);

    const float ki = kbuf[cur][i];
    red[i] = ki * ki;
    __syncthreads();
    float k2 = 0.0f;
    #pragma unroll
    for (int j = 0; j < HD; ++j) k2 += red[j];
    __syncthreads();                       // all reads of red done
    const float kn = ki / fmaxf(sqrtf(k2), 1e-12f);
    kn_s[i] = kn;
    __syncthreads();
    float vp = 0.0f;
    #pragma unroll
    for (int j = 0; j < HD; ++j) vp += M[j] * kn_s[j];
    const float err = ki - vp;
    red[i] = err * err;
    __syncthreads();
    float e2 = 0.0f;
    #pragma unroll
    for (int j = 0; j < HD; ++j) e2 += red[j];
    // gate: ||err|| >= 0.4*||k||  <=>  e2 >= 0.16*k2
    const float ge = (e2 >= 0.16f * k2) ? err : 0.0f;
    #pragma unroll
    for (int j = 0; j < HD; ++j) M[j] += ge * kn_s[j];

    wait_async0();                         // prefetch landed
    __syncthreads();                       // protect kn_s/red/kbuf for next step
  }

  // read = M @ q  (q = row L-1, already staged in kbuf[(L-1)&1])
  kn_s[i] = kbuf[(LSEQ - 1) & 1][i];
  __syncthreads();
  float rd = 0.0f;
  #pragma unroll
  for (int j = 0; j < HD; ++j) rd += M[j] * kn_s[j];
  __syncthreads();
  red[i] = rd;
  __syncthreads();

  // proj = read @ rpW + rpb
  float p = rpb[i];
  #pragma unroll
  for (int j = 0; j < HD; ++j) p += red[j] * rpW[j * HD + i];
  proj[b * HD + i] = p;
}

// ---------------------------------------------------------------------------
// Kernel 3: out = proj(64x64) @ outW(64x32000) + outb, fp32 WMMA tiles.
// One wave per 16x16 output tile; grid (V/16, B/16). Memory bound:
// ~16 MB outW/out traffic at 23.3 TB/s; prefetch keeps loads streaming.
// ---------------------------------------------------------------------------
__global__ __launch_bounds__(32) void out_gemm_kernel(
    const float* __restrict__ proj, const float* __restrict__ outW,
    const float* __restrict__ outb, float* __restrict__ out)
{
  const int lane = threadIdx.x;
  const int half = lane >> 4;
  const int l16  = lane & 15;
  const int n0 = blockIdx.x * 16;
  const int m0 = blockIdx.y * 16;

  const float bias = outb[n0 + l16];
  v8f acc;
  #pragma unroll
  for (int v = 0; v < 8; ++v) acc[v] = bias;

  #pragma unroll
  for (int k0 = 0; k0 < HD; k0 += 4) {
    const int ka = k0 + 2 * half;
    if (k0 + 4 < HD)  // emits global_prefetch_b8
      __builtin_prefetch(&outW[(size_t)(ka + 4) * VOC + n0 + l16], 0, 1);
    v2f a, b;
    a.x = proj[(m0 + l16) * HD + ka];
    a.y = proj[(m0 + l16) * HD + ka + 1];
    b.x = outW[(size_t)ka * VOC + n0 + l16];
    b.y = outW[(size_t)(ka + 1) * VOC + n0 + l16];
    acc = wmma_f32x4(a, b, acc);
  }
  #pragma unroll
  for (int v = 0; v < 8; ++v)
    out[(size_t)(m0 + v + 8 * half) * VOC + n0 + l16] = acc[v];
}

// ---------------------------------------------------------------------------
extern "C" void kernel_launch(void* const* d_in, const int* in_sizes, int n_in,
                              void* d_out, int out_size, void* d_ws, size_t ws_size,
                              hipStream_t stream) {
  const int*   seq   = (const int*)  d_in[0];
  const float* embed = (const float*)d_in[1];
  const float* W1    = (const float*)d_in[2];
  const float* b1    = (const float*)d_in[3];
  const float* W2    = (const float*)d_in[4];
  const float* b2    = (const float*)d_in[5];
  const float* gamma = (const float*)d_in[6];
  const float* beta  = (const float*)d_in[7];
  const float* kpW   = (const float*)d_in[8];
  const float* rpW   = (const float*)d_in[9];
  const float* rpb   = (const float*)d_in[10];
  const float* outW  = (const float*)d_in[11];
  const float* outb  = (const float*)d_in[12];
  float* out = (float*)d_out;

  float* k_all = (float*)d_ws;                            // B*L*H = 16 MB
  float* proj  = k_all + (size_t)BATCH * LSEQ * HD;       // B*H

  token_prep_kernel<<<(BATCH * LSEQ) / 16, 32, 0, stream>>>(
      seq, embed, W1, b1, W2, b2, gamma, beta, kpW, k_all);
  scan_kernel<<<BATCH, 64, 0, stream>>>(k_all, rpW, rpb, proj);
  out_gemm_kernel<<<dim3(VOC / 16, BATCH / 16), 32, 0, stream>>>(
      proj, outW, outb, out);
}